// RootedGraphPredictor_16681652977790
// MI455X (gfx1250) — compile-verified
//
#include <hip/hip_runtime.h>
#include <hip/hip_bf16.h>
#include <stdint.h>

// Problem constants (from the reference): B=4, L=2048, E=512, D=128
#define BB 4
#define LL 2048
#define EE 512
#define DD 128

typedef float v2f __attribute__((ext_vector_type(2)));
typedef float v8f __attribute__((ext_vector_type(8)));

// D = A(16x4 f32) * B(4x16 f32) + C(16x16 f32), wave32, one 16x16 tile per wave.
__device__ __forceinline__ v8f wmma_f32_4(v2f a, v2f b, v8f c) {
    return __builtin_amdgcn_wmma_f32_16x16x4_f32(
        /*neg_a=*/false, a, /*neg_b=*/false, b,
        /*c_mod=*/(short)0, c, /*reuse_a=*/false, /*reuse_b=*/false);
}

// CDNA5 async global->LDS copy (ASYNCcnt-tracked, no VGPR data movement).
// LDS destination = per-lane byte offset (low 32 bits of the flat pointer:
// the LDS aperture maps addr[31:0] directly to the LDS address).
__device__ __forceinline__ void async_load_b128(const void* gaddr, void* lds_ptr) {
    uint32_t lds_off = (uint32_t)(uintptr_t)lds_ptr;
    asm volatile("global_load_async_to_lds_b128 %0, %1, off"
                 :: "v"(lds_off), "v"(gaddr) : "memory");
}
__device__ __forceinline__ void wait_async_all() {
    asm volatile("s_wait_asynccnt 0x0" ::: "memory");
}

// ---------------------------------------------------------------------------
// Kernel 1: K-paired transpose of W [D,E] -> Wp, where
//   Wp[(e>>1)*(2*D) + d*2 + (e&1)] = W[d][e]
// so a WMMA B-fragment (two consecutive K values, one N column) is one
// contiguous 8-byte load. 32x32 LDS tile transpose, blockDim (32,8).
// ---------------------------------------------------------------------------
__global__ __launch_bounds__(256) void wt_transpose_kernel(
    const float* __restrict__ W, float* __restrict__ Wp) {
    __shared__ float tile[32][33];
    const int e0 = blockIdx.x * 32;
    const int d0 = blockIdx.y * 32;
    const int tx = threadIdx.x;   // 0..31
    const int ty = threadIdx.y;   // 0..7
    #pragma unroll
    for (int i = 0; i < 32; i += 8)
        tile[ty + i][tx] = W[(size_t)(d0 + ty + i) * EE + e0 + tx];
    __syncthreads();
    #pragma unroll
    for (int i = 0; i < 32; i += 8) {
        const int e = e0 + ty + i;
        const int d = d0 + tx;
        Wp[(size_t)(e >> 1) * (2 * DD) + d * 2 + (e & 1)] = tile[tx][ty + i];
    }
}

// ---------------------------------------------------------------------------
// Kernel 2: dep[m][n] = sum_k emb[m][k] * W[n][k],  m over B*L, n over D.
// Block = 256 threads = 8 waves; block owns 16 rows x all 128 cols (wave w ->
// N tile w). A tile (16x512 f32) staged via async global->LDS, row stride
// 516 floats (conflict-free banks, 16B-aligned rows). Two independent
// accumulator chains (even/odd K-step) so the XDL pipe is never RAW-stalled.
// ---------------------------------------------------------------------------
#define A_STRIDE 516
__global__ __launch_bounds__(256) void dep_gemm_kernel(
    const float* __restrict__ emb, const float* __restrict__ Wp,
    float* __restrict__ dep) {
    __shared__ float As[16 * A_STRIDE];
    const int tid = threadIdx.x;
    const int m0  = blockIdx.x * 16;              // row tile over B*L = 8192

    // Async stage: 16 rows x 512 floats = 2048 float4, 8 per thread.
    #pragma unroll
    for (int i = 0; i < 8; ++i) {
        int idx = tid + i * 256;                  // float4 index
        int r = idx >> 7;                         // /128 float4-per-row
        int q = idx & 127;
        async_load_b128(emb + (size_t)(m0 + r) * EE + q * 4,
                        &As[r * A_STRIDE + q * 4]);
    }
    wait_async_all();
    __syncthreads();

    const int wave = tid >> 5;
    const int lane = tid & 31;
    const int row  = lane & 15;                   // M (for A) / N (for B)
    const int hi   = lane >> 4;                   // lane-half selects K pair
    const int n0   = wave * 16;

    const float* arow = &As[row * A_STRIDE];
    const float* bcol = Wp + (size_t)(n0 + row) * 2;   // + (k>>1)*256
    v8f ca = {}, cb = {};
    for (int k0 = 0; k0 < EE; k0 += 8) {
        const int kb0 = k0 + (hi << 1);           // K0/K2 by lane-half
        const int kb1 = kb0 + 4;
        v2f a0, a1;
        a0.x = arow[kb0]; a0.y = arow[kb0 + 1];
        a1.x = arow[kb1]; a1.y = arow[kb1 + 1];
        const v2f b0 = *(const v2f*)(bcol + (size_t)(kb0 >> 1) * (2 * DD));
        const v2f b1 = *(const v2f*)(bcol + (size_t)(kb1 >> 1) * (2 * DD));
        ca = wmma_f32_4(a0, b0, ca);
        cb = wmma_f32_4(a1, b1, cb);
    }

    // C/D layout: VGPR r -> M = r + hi*8, N = lane&15.
    float* out = dep + (size_t)m0 * DD + n0;
    #pragma unroll
    for (int r = 0; r < 8; ++r)
        out[(size_t)(r + hi * 8) * DD + row] = ca[r] + cb[r];
}

// ---------------------------------------------------------------------------
// Kernel 3: sq[row] = ||dep[row]||^2 over D=128. One wave32 per row:
// each lane float4-loads 4 elements, butterfly-reduce across the wave.
// ---------------------------------------------------------------------------
__global__ __launch_bounds__(256) void row_sq_kernel(
    const float* __restrict__ dep, float* __restrict__ sq) {
    const int grow = (blockIdx.x * 256 + threadIdx.x) >> 5;   // 0..B*L-1
    const int lane = threadIdx.x & 31;
    const float4 v = *(const float4*)(dep + (size_t)grow * DD + lane * 4);
    float s = v.x * v.x + v.y * v.y + v.z * v.z + v.w * v.w;
    #pragma unroll
    for (int off = 16; off > 0; off >>= 1)
        s += __shfl_xor(s, off, 32);
    if (lane == 0) sq[grow] = s;
}

// ---------------------------------------------------------------------------
// Kernel 4: distances[b][i][j] = sq[i] + sq[j] - 2 * dot(dep_i, dep_j).
// Block = 256 threads = 8 waves computes one 128(i) x 64(j) tile. dep_i
// (128x128 f32) and dep_j (64x128 f32) staged via async global->LDS, row
// stride 132 floats (conflict-free: bank = (4*row+k)%64, 16B-aligned rows).
// Wave w owns a 2x2 grid of 16x16 subtiles (si=2*(w>>1), sj=2*(w&1)):
// per K-step, 2 A + 2 B fragment loads feed 4 WMMAs into 4 independent
// accumulator chains (4 wmma : 2 ds_load_2addr_b64).
// ---------------------------------------------------------------------------
#define T_STRIDE 132
__global__ __launch_bounds__(256) void dist_kernel(
    const float* __restrict__ dep, const float* __restrict__ sq,
    float* __restrict__ dist) {
    __shared__ float Ai[128 * T_STRIDE];   // 67,584 B
    __shared__ float Bj[64 * T_STRIDE];    // 33,792 B
    const int b   = blockIdx.z;
    const int i0  = blockIdx.y * 128;
    const int j0  = blockIdx.x * 64;
    const int tid = threadIdx.x;
    const float* depb = dep + (size_t)b * LL * DD;

    // Async stage: Ai = 128 rows (16 float4/thread), Bj = 64 rows (8/thread).
    #pragma unroll
    for (int i = 0; i < 16; ++i) {
        int idx = tid + i * 256;
        int r = idx >> 5;                          // /32 float4-per-row
        int q = idx & 31;
        async_load_b128(depb + (size_t)(i0 + r) * DD + q * 4,
                        &Ai[r * T_STRIDE + q * 4]);
    }
    #pragma unroll
    for (int i = 0; i < 8; ++i) {
        int idx = tid + i * 256;
        int r = idx >> 5;
        int q = idx & 31;
        async_load_b128(depb + (size_t)(j0 + r) * DD + q * 4,
                        &Bj[r * T_STRIDE + q * 4]);
    }
    wait_async_all();
    __syncthreads();

    const int wave = tid >> 5;
    const int lane = tid & 31;
    const int row  = lane & 15;
    const int hi   = lane >> 4;
    const int si0  = (wave >> 1) * 2;              // i subtiles si0, si0+1 (of 8)
    const int sj0  = (wave & 1) * 2;               // j subtiles sj0, sj0+1 (of 4)

    const float* arow0 = &Ai[(si0 * 16 + row) * T_STRIDE];
    const float* arow1 = &Ai[((si0 + 1) * 16 + row) * T_STRIDE];
    const float* brow0 = &Bj[(sj0 * 16 + row) * T_STRIDE];
    const float* brow1 = &Bj[((sj0 + 1) * 16 + row) * T_STRIDE];

    v8f c00 = {}, c01 = {}, c10 = {}, c11 = {};
    for (int k0 = 0; k0 < DD; k0 += 4) {
        const int kb = k0 + (hi << 1);
        v2f a0, a1, b0, b1;
        a0.x = arow0[kb]; a0.y = arow0[kb + 1];
        a1.x = arow1[kb]; a1.y = arow1[kb + 1];
        b0.x = brow0[kb]; b0.y = brow0[kb + 1];
        b1.x = brow1[kb]; b1.y = brow1[kb + 1];
        c00 = wmma_f32_4(a0, b0, c00);
        c01 = wmma_f32_4(a0, b1, c01);
        c10 = wmma_f32_4(a1, b0, c10);
        c11 = wmma_f32_4(a1, b1, c11);
    }

    // Epilogue: dist = sq_i + sq_j - 2*gram.  sq_i rows via aligned b128
    // loads (ibase* are multiples of 8 -> 32B aligned).
    const float* sqb   = sq + b * LL;
    const float  sqj0  = sqb[j0 + sj0 * 16 + row];
    const float  sqj1  = sqb[j0 + (sj0 + 1) * 16 + row];
    const int    ib0   = i0 + si0 * 16 + hi * 8;
    const int    ib1   = i0 + (si0 + 1) * 16 + hi * 8;
    const float4 s0a = *(const float4*)(sqb + ib0);
    const float4 s0b = *(const float4*)(sqb + ib0 + 4);
    const float4 s1a = *(const float4*)(sqb + ib1);
    const float4 s1b = *(const float4*)(sqb + ib1 + 4);
    const float si_lo[8] = {s0a.x, s0a.y, s0a.z, s0a.w, s0b.x, s0b.y, s0b.z, s0b.w};
    const float si_hi[8] = {s1a.x, s1a.y, s1a.z, s1a.w, s1b.x, s1b.y, s1b.z, s1b.w};

    float* outb = dist + (size_t)b * LL * LL;
    #pragma unroll
    for (int r = 0; r < 8; ++r) {
        float* orow0 = outb + (size_t)(ib0 + r) * LL + j0;
        float* orow1 = outb + (size_t)(ib1 + r) * LL + j0;
        orow0[sj0 * 16 + row]       = fmaf(-2.0f, c00[r], si_lo[r] + sqj0);
        orow0[(sj0 + 1) * 16 + row] = fmaf(-2.0f, c01[r], si_lo[r] + sqj1);
        orow1[sj0 * 16 + row]       = fmaf(-2.0f, c10[r], si_hi[r] + sqj0);
        orow1[(sj0 + 1) * 16 + row] = fmaf(-2.0f, c11[r], si_hi[r] + sqj1);
    }
}

// ---------------------------------------------------------------------------
// d_in[0] = emb [B,L,E] f32; d_in[1] = W [D,E] f32.
// d_out = dep [B,L,D] followed by distances [B,L,L] (f32, concatenated).
// d_ws  = Wp (E*D floats, 256KB) + sq (B*L floats, 32KB).
// ---------------------------------------------------------------------------
extern "C" void kernel_launch(void* const* d_in, const int* in_sizes, int n_in,
                              void* d_out, int out_size, void* d_ws, size_t ws_size,
                              hipStream_t stream) {
    const float* emb = (const float*)d_in[0];
    const float* W   = (const float*)d_in[1];
    float* out  = (float*)d_out;
    float* dep  = out;                                   // B*L*D
    float* dist = out + (size_t)BB * LL * DD;            // B*L*L
    float* Wp   = (float*)d_ws;                          // E*D (K-paired)
    float* sq   = Wp + (size_t)EE * DD;                  // B*L

    wt_transpose_kernel<<<dim3(EE / 32, DD / 32), dim3(32, 8), 0, stream>>>(W, Wp);
    dep_gemm_kernel<<<dim3((BB * LL) / 16), dim3(256), 0, stream>>>(emb, Wp, dep);
    row_sq_kernel<<<dim3((BB * LL * 32) / 256), dim3(256), 0, stream>>>(dep, sq);
    dist_kernel<<<dim3(LL / 64, LL / 128, BB), dim3(256), 0, stream>>>(dep, sq, dist);
}